// GAT_82497731821610
// MI455X (gfx1250) — compile-verified
//
#include <hip/hip_runtime.h>
#include <math.h>

#define NNODES 50000
#define NEDGES 800000
#define HEADS 4
#define C1 16
#define C2 8
#define FIN 128
#define F1 (HEADS * C1) /* 64 */
#define F2 (HEADS * C2) /* 32 */
#define ETOT (NEDGES + NNODES)
#define NEG_SLOPE 0.2f
#define GAT_EPS 1e-16f

typedef __attribute__((ext_vector_type(2))) float v2f;
typedef __attribute__((ext_vector_type(8))) float v8f;
typedef int v4i __attribute__((vector_size(16)));

// gfx1250 async Global->LDS (ASYNCcnt-tracked) if the toolchain exposes it.
#if defined(__AMDGCN__) &&                                         \
    __has_builtin(__builtin_amdgcn_global_load_async_to_lds_b128) && \
    __has_builtin(__builtin_amdgcn_s_wait_asynccnt)
#define HAVE_ASYNC_LDS 1
#else
#define HAVE_ASYNC_LDS 0
#endif

// ---------------------------------------------------------------------------
// Y[nrows x FOUT] = X[nrows x KDIM] @ W[KDIM x FOUT], f32 WMMA (16x16x4).
// W (<=32KB) is staged once per block into LDS via async Global->LDS DMA,
// then each wave computes a 16-row x FOUT tile.
// A frag (16x4 f32): lane half (lane>>4) picks K pair, vgpr picks K in pair,
//                    lane&15 = M.
// B frag (4x16 f32): same K mapping, lane&15 = N.
// D frag (16x16 f32): vgpr v, lane l -> M = v + 8*(l>>4), N = l&15.
// ---------------------------------------------------------------------------
template <int NTILES, int KDIM>
__global__ void gemm_wmma_f32(const float* __restrict__ X,
                              const float* __restrict__ W,
                              float* __restrict__ Y,
                              int nrows) {
  constexpr int FOUT = NTILES * 16;
  constexpr int WELEMS = KDIM * FOUT;
  __shared__ __align__(16) float wlds[WELEMS];

  // ---- cooperative staging of W into LDS (before any wave-level exit) ----
#if HAVE_ASYNC_LDS
  for (int t = (int)threadIdx.x * 4; t < WELEMS; t += 256 * 4) {
    __builtin_amdgcn_global_load_async_to_lds_b128(
        (__attribute__((address_space(1))) v4i*)(float*)(W + t),
        (__attribute__((address_space(3))) v4i*)(wlds + t), 0, 0);
  }
  __builtin_amdgcn_s_wait_asynccnt(0);
#else
  for (int t = (int)threadIdx.x * 4; t < WELEMS; t += 256 * 4) {
    *(float4*)(wlds + t) = *(const float4*)(W + t);
  }
#endif
  __syncthreads();

  int wave = (int)((blockIdx.x * blockDim.x + threadIdx.x) >> 5);
  int lane = threadIdx.x & 31;
  int row0 = wave * 16;
  if (row0 < nrows) {  // uniform per wave: EXEC stays all-1s for WMMA
    int m = lane & 15;
    int half = lane >> 4;

    v8f acc[NTILES];
#pragma unroll
    for (int j = 0; j < NTILES; ++j) {
#pragma unroll
      for (int e = 0; e < 8; ++e) acc[j][e] = 0.0f;
    }

    const float* xrow = X + (size_t)(row0 + m) * KDIM + half * 2;
    for (int k0 = 0; k0 < KDIM; k0 += 4) {
      v2f a;
      a.x = xrow[k0];
      a.y = xrow[k0 + 1];
      int kk = k0 + half * 2;
#pragma unroll
      for (int j = 0; j < NTILES; ++j) {
        int n = j * 16 + m;
        v2f b;
        b.x = wlds[kk * FOUT + n];
        b.y = wlds[(kk + 1) * FOUT + n];
        acc[j] = __builtin_amdgcn_wmma_f32_16x16x4_f32(
            /*neg_a=*/false, a, /*neg_b=*/false, b,
            /*c_mod=*/(short)0, acc[j], /*reuse_a=*/false, /*reuse_b=*/false);
      }
    }

#pragma unroll
    for (int j = 0; j < NTILES; ++j) {
      int n = j * 16 + m;
#pragma unroll
      for (int v = 0; v < 8; ++v) {
        int row = row0 + v + half * 8;
        Y[(size_t)row * FOUT + n] = acc[j][v];
      }
    }
  }
}

// ---------------------------------------------------------------------------
__global__ void fill_kernel(float* __restrict__ p, float v, int n) {
  int i = blockIdx.x * blockDim.x + threadIdx.x;
  if (i < n) p[i] = v;
}

// Per-(node, head) attention logits: als/ald laid out [N, H].
template <int C>
__global__ void node_logits(const float* __restrict__ h,
                            const float* __restrict__ a_src,
                            const float* __restrict__ a_dst,
                            float* __restrict__ als,
                            float* __restrict__ ald) {
  int i = blockIdx.x * blockDim.x + threadIdx.x;
  if (i >= NNODES * HEADS) return;
  int node = i / HEADS;
  int hd = i % HEADS;
  const float* hp = h + (size_t)node * (HEADS * C) + hd * C;
  float ss = 0.f, sd = 0.f;
#pragma unroll
  for (int c = 0; c < C; ++c) {
    float v = hp[c];
    ss += v * a_src[hd * C + c];
    sd += v * a_dst[hd * C + c];
  }
  als[i] = ss;
  ald[i] = sd;
}

__device__ __forceinline__ void edge_endpoints(const int* __restrict__ ei,
                                               int e, int& s, int& d) {
  if (e < NEDGES) {
    s = ei[e];
    d = ei[NEDGES + e];
  } else {
    s = d = e - NEDGES;  // self-loop
  }
}

__device__ __forceinline__ float edge_logit(const float* __restrict__ als,
                                            const float* __restrict__ ald,
                                            int s, int d, int hd) {
  float v = als[s * HEADS + hd] + ald[d * HEADS + hd];
  return v >= 0.f ? v : NEG_SLOPE * v;  // leaky relu
}

// Segment max over destinations (native f32 atomic max -> global_atomic_max_num_f32)
__global__ void edge_max_kernel(const int* __restrict__ ei,
                                const float* __restrict__ als,
                                const float* __restrict__ ald,
                                float* __restrict__ mx) {
  int idx = blockIdx.x * blockDim.x + threadIdx.x;
  if (idx >= ETOT * HEADS) return;
  int e = idx / HEADS, hd = idx % HEADS;
  int s, d;
  edge_endpoints(ei, e, s, d);
  float v = edge_logit(als, ald, s, d, hd);
  atomicMax(&mx[d * HEADS + hd], v);
}

// p[idx] = exp(e - m[dst]); segment sum via atomicAdd
__global__ void edge_expsum_kernel(const int* __restrict__ ei,
                                   const float* __restrict__ als,
                                   const float* __restrict__ ald,
                                   const float* __restrict__ mx,
                                   float* __restrict__ sm,
                                   float* __restrict__ p) {
  int idx = blockIdx.x * blockDim.x + threadIdx.x;
  if (idx >= ETOT * HEADS) return;
  int e = idx / HEADS, hd = idx % HEADS;
  int s, d;
  edge_endpoints(ei, e, s, d);
  float v = edge_logit(als, ald, s, d, hd);
  float pe = __expf(v - mx[d * HEADS + hd]);
  p[idx] = pe;
  atomicAdd(&sm[d * HEADS + hd], pe);
}

// out[dst,h,:] += alpha * h[src,h,:]
template <int C>
__global__ void edge_aggregate(const int* __restrict__ ei,
                               const float* __restrict__ h,
                               const float* __restrict__ p,
                               const float* __restrict__ sm,
                               float* __restrict__ out) {
  int idx = blockIdx.x * blockDim.x + threadIdx.x;
  if (idx >= ETOT * HEADS) return;
  int e = idx / HEADS, hd = idx % HEADS;
  int s, d;
  edge_endpoints(ei, e, s, d);
  float alpha = p[idx] / (sm[d * HEADS + hd] + GAT_EPS);
  const float* hs = h + (size_t)s * (HEADS * C) + hd * C;
  float* od = out + (size_t)d * (HEADS * C) + hd * C;
#pragma unroll
  for (int c = 0; c < C; ++c) atomicAdd(&od[c], alpha * hs[c]);
}

__global__ void bias_relu_kernel(float* __restrict__ y,
                                 const float* __restrict__ bias,
                                 int total, int F) {
  int i = blockIdx.x * blockDim.x + threadIdx.x;
  if (i >= total) return;
  float v = y[i] + bias[i % F];
  y[i] = v > 0.f ? v : 0.f;
}

// Fused bias add + log_softmax over the 32 output channels.
__global__ void logsoftmax_out_kernel(const float* __restrict__ agg,
                                      const float* __restrict__ bias,
                                      float* __restrict__ out) {
  int i = blockIdx.x * blockDim.x + threadIdx.x;
  if (i >= NNODES) return;
  float v[F2];
  float mx = -3.4e38f;
#pragma unroll
  for (int c = 0; c < F2; ++c) {
    v[c] = agg[(size_t)i * F2 + c] + bias[c];
    mx = fmaxf(mx, v[c]);
  }
  float ssum = 0.f;
#pragma unroll
  for (int c = 0; c < F2; ++c) ssum += __expf(v[c] - mx);
  float ls = __logf(ssum);
#pragma unroll
  for (int c = 0; c < F2; ++c) out[(size_t)i * F2 + c] = v[c] - mx - ls;
}

// ---------------------------------------------------------------------------
static inline int nblk(long long n, int tb) { return (int)((n + tb - 1) / tb); }

extern "C" void kernel_launch(void* const* d_in, const int* in_sizes, int n_in,
                              void* d_out, int out_size, void* d_ws,
                              size_t ws_size, hipStream_t stream) {
  (void)in_sizes; (void)n_in; (void)out_size; (void)ws_size;

  const float* x   = (const float*)d_in[0];
  const int*   ei  = (const int*)d_in[1];  // [2, E] int32
  const float* W1  = (const float*)d_in[2];
  const float* as1 = (const float*)d_in[3];
  const float* ad1 = (const float*)d_in[4];
  const float* b1  = (const float*)d_in[5];
  const float* W2  = (const float*)d_in[6];
  const float* as2 = (const float*)d_in[7];
  const float* ad2 = (const float*)d_in[8];
  const float* b2  = (const float*)d_in[9];
  float* out = (float*)d_out;

  // workspace layout (floats)
  float* ws   = (float*)d_ws;
  float* h1   = ws;                                   // N*F1
  float* agg1 = h1 + (size_t)NNODES * F1;             // N*F1 (layer-2 input after relu)
  float* h2   = agg1 + (size_t)NNODES * F1;           // N*F2
  float* agg2 = h2 + (size_t)NNODES * F2;             // N*F2
  float* als  = agg2 + (size_t)NNODES * F2;           // N*H
  float* ald  = als + (size_t)NNODES * HEADS;         // N*H
  float* mx   = ald + (size_t)NNODES * HEADS;         // N*H
  float* sm   = mx + (size_t)NNODES * HEADS;          // N*H
  float* p    = sm + (size_t)NNODES * HEADS;          // ETOT*H

  const int TB = 256;
  const long long NH = (long long)NNODES * HEADS;
  const long long EH = (long long)ETOT * HEADS;
  const int gemm_threads = (NNODES / 16) * 32;  // one wave per 16-row tile

  // ---------------- layer 1 ----------------
  fill_kernel<<<nblk(NH, TB), TB, 0, stream>>>(mx, -INFINITY, (int)NH);
  fill_kernel<<<nblk(NH, TB), TB, 0, stream>>>(sm, 0.0f, (int)NH);
  fill_kernel<<<nblk((long long)NNODES * F1, TB), TB, 0, stream>>>(
      agg1, 0.0f, NNODES * F1);

  gemm_wmma_f32<4, FIN><<<nblk(gemm_threads, TB), TB, 0, stream>>>(
      x, W1, h1, NNODES);
  node_logits<C1><<<nblk(NH, TB), TB, 0, stream>>>(h1, as1, ad1, als, ald);
  edge_max_kernel<<<nblk(EH, TB), TB, 0, stream>>>(ei, als, ald, mx);
  edge_expsum_kernel<<<nblk(EH, TB), TB, 0, stream>>>(ei, als, ald, mx, sm, p);
  edge_aggregate<C1><<<nblk(EH, TB), TB, 0, stream>>>(ei, h1, p, sm, agg1);
  bias_relu_kernel<<<nblk((long long)NNODES * F1, TB), TB, 0, stream>>>(
      agg1, b1, NNODES * F1, F1);

  // ---------------- layer 2 ----------------
  fill_kernel<<<nblk(NH, TB), TB, 0, stream>>>(mx, -INFINITY, (int)NH);
  fill_kernel<<<nblk(NH, TB), TB, 0, stream>>>(sm, 0.0f, (int)NH);
  fill_kernel<<<nblk((long long)NNODES * F2, TB), TB, 0, stream>>>(
      agg2, 0.0f, NNODES * F2);

  gemm_wmma_f32<2, F1><<<nblk(gemm_threads, TB), TB, 0, stream>>>(
      agg1, W2, h2, NNODES);
  node_logits<C2><<<nblk(NH, TB), TB, 0, stream>>>(h2, as2, ad2, als, ald);
  edge_max_kernel<<<nblk(EH, TB), TB, 0, stream>>>(ei, als, ald, mx);
  edge_expsum_kernel<<<nblk(EH, TB), TB, 0, stream>>>(ei, als, ald, mx, sm, p);
  edge_aggregate<C2><<<nblk(EH, TB), TB, 0, stream>>>(ei, h2, p, sm, agg2);

  logsoftmax_out_kernel<<<nblk(NNODES, TB), TB, 0, stream>>>(agg2, b2, out);
}